// TopKRouter_38628935860428
// MI455X (gfx1250) — compile-verified
//
#include <hip/hip_runtime.h>
#include <hip/hip_bf16.h>
#include <math.h>

typedef float v2f __attribute__((ext_vector_type(2)));
typedef float v8f __attribute__((ext_vector_type(8)));

#define DIM      4096
#define N_EXP    64
#define TOKENS   16384
#define ROWS_WG  128          // tokens per workgroup (8 waves x 16 rows)
#define LDS_PITCH 65          // 64 + 1 pad -> conflict-free column reads

__global__ __launch_bounds__(256)
void topk_router_kernel(const float* __restrict__ x,
                        const float* __restrict__ w_gate,
                        float* __restrict__ gates,
                        float* __restrict__ topk_vals,
                        float* __restrict__ topk_idx) {
    __shared__ float lds[ROWS_WG * LDS_PITCH];

    const int lane   = threadIdx.x & 31;
    const int wv     = threadIdx.x >> 5;      // wave id 0..7
    const int laneM  = lane & 15;             // row (A) / expert-in-tile (B)
    const int laneHi = lane >> 4;             // K pair select
    const int m0     = blockIdx.x * ROWS_WG + wv * 16;

    // A: 16 tokens x 4 K per step.  B: 4 K x 16 experts, 4 tiles cover 64 experts.
    const float* aptr = x      + (size_t)(m0 + laneM) * DIM + 2 * laneHi;
    const float* bptr = w_gate + (size_t)laneM        * DIM + 2 * laneHi;

    v8f acc0 = {}, acc1 = {}, acc2 = {}, acc3 = {};

    #pragma unroll 8
    for (int k = 0; k < DIM; k += 4) {
        // x is streamed exactly once -> non-temporal, keep WGP$ for w_gate
        v2f a  = __builtin_nontemporal_load((const v2f*)(aptr + k));
        v2f b0 = *(const v2f*)(bptr + k);
        v2f b1 = *(const v2f*)(bptr + 16 * DIM + k);
        v2f b2 = *(const v2f*)(bptr + 32 * DIM + k);
        v2f b3 = *(const v2f*)(bptr + 48 * DIM + k);
        // 8 args: (neg_a, A, neg_b, B, c_mod, C, reuse_a, reuse_b)
        acc0 = __builtin_amdgcn_wmma_f32_16x16x4_f32(false, a, false, b0, (short)0, acc0, false, false);
        acc1 = __builtin_amdgcn_wmma_f32_16x16x4_f32(false, a, false, b1, (short)0, acc1, false, false);
        acc2 = __builtin_amdgcn_wmma_f32_16x16x4_f32(false, a, false, b2, (short)0, acc2, false, false);
        acc3 = __builtin_amdgcn_wmma_f32_16x16x4_f32(false, a, false, b3, (short)0, acc3, false, false);
    }

    // Spill logits tile to LDS.  C/D layout: VGPR r, lanes 0-15 -> M=r, lanes 16-31 -> M=r+8.
    #pragma unroll
    for (int r = 0; r < 8; ++r) {
        const int M    = r + (laneHi ? 8 : 0);
        const int base = (wv * 16 + M) * LDS_PITCH + laneM;
        lds[base +  0] = acc0[r];
        lds[base + 16] = acc1[r];
        lds[base + 32] = acc2[r];
        lds[base + 48] = acc3[r];
    }
    __syncthreads();

    // One thread per token: softmax over 64 experts + top-2.
    const int row = threadIdx.x;
    if (row < ROWS_WG) {
        const float* l = &lds[row * LDS_PITCH];
        float m = -INFINITY;
        #pragma unroll
        for (int e = 0; e < N_EXP; ++e) m = fmaxf(m, l[e]);
        float s = 0.f;
        #pragma unroll
        for (int e = 0; e < N_EXP; ++e) s += __expf(l[e] - m);
        const float inv = 1.f / s;

        const int token = blockIdx.x * ROWS_WG + row;
        float* grow = gates + (size_t)token * N_EXP;
        float v0 = -INFINITY, v1 = -INFINITY;
        int   i0 = 0, i1 = 0;
        #pragma unroll
        for (int e = 0; e < N_EXP; ++e) {
            const float g = __expf(l[e] - m) * inv;
            __builtin_nontemporal_store(g, &grow[e]);   // write-once output
            if (g > v0)      { v1 = v0; i1 = i0; v0 = g; i0 = e; }
            else if (g > v1) { v1 = g;  i1 = e; }
        }
        topk_vals[token * 2 + 0] = v0;
        topk_vals[token * 2 + 1] = v1;
        topk_idx [token * 2 + 0] = (float)i0;
        topk_idx [token * 2 + 1] = (float)i1;
    }
}

extern "C" void kernel_launch(void* const* d_in, const int* in_sizes, int n_in,
                              void* d_out, int out_size, void* d_ws, size_t ws_size,
                              hipStream_t stream) {
    (void)in_sizes; (void)n_in; (void)out_size; (void)d_ws; (void)ws_size;
    const float* x      = (const float*)d_in[0];
    const float* w_gate = (const float*)d_in[1];

    float* gates     = (float*)d_out;                          // [TOKENS, 64]
    float* topk_vals = gates + (size_t)TOKENS * N_EXP;         // [TOKENS, 2]
    float* topk_idx  = topk_vals + (size_t)TOKENS * 2;         // [TOKENS, 2] as float

    dim3 grid(TOKENS / ROWS_WG);   // 128 workgroups
    dim3 block(256);               // 8 waves (wave32)
    topk_router_kernel<<<grid, block, 0, stream>>>(x, w_gate, gates, topk_vals, topk_idx);
}